// DensePropMaxPool_6004364280204
// MI455X (gfx1250) — compile-verified
//
#include <hip/hip_runtime.h>
#include <hip/hip_bf16.h>

typedef __attribute__((ext_vector_type(4))) float f4;

#define NB 32      // B
#define NH 512     // H
#define NC 64      // C
#define NP 1024    // P

// d_out layout (floats): [props_h^T (B,P,H)] [map_h (B,H,C,C)] [map_mask (B,1,C,C)]
#define OUT2_OFF (NB * NP * NH)                   // 16,777,216
#define OUT3_OFF (OUT2_OFF + NB * NH * NC * NC)   // 83,886,080

// ---------------------------------------------------------------------------
// Kernel 1: map_h[b,h,i,j] = max(x[b,h,i..j]) for j>=i else 0 ; mask = (j>=i).
// One 16-lane subgroup per (b,h) row; 4 columns per lane (float4 NT stores).
// x row staged via CDNA5 async global->LDS copy (ASYNCcnt path).
// ---------------------------------------------------------------------------
__global__ __launch_bounds__(64)
void dpm_map_kernel(const float* __restrict__ x,
                    float* __restrict__ map_h,
                    float* __restrict__ map_mask) {
    __shared__ float xs[4 * NC];
    const int t   = threadIdx.x;
    const int s   = t >> 4;          // sub-row 0..3
    const int j0  = (t & 15) * 4;    // first of 4 owned columns
    const int row = blockIdx.x * 4 + s;   // flat b*H + h

    const float* gp = x + (size_t)row * NC + j0;
#if defined(__gfx1250__)
    {
        // low 32 bits of a generic shared pointer == LDS byte address
        unsigned la = (unsigned)(unsigned long long)(const void*)&xs[s * NC + j0];
        asm volatile("global_load_async_to_lds_b128 %0, %1, off"
                     :: "v"(la), "v"(gp) : "memory");
        asm volatile("s_wait_asynccnt 0" ::: "memory");
    }
#else
    xs[s * NC + j0 + 0] = gp[0];
    xs[s * NC + j0 + 1] = gp[1];
    xs[s * NC + j0 + 2] = gp[2];
    xs[s * NC + j0 + 3] = gp[3];
#endif
    __syncthreads();

    f4 m = {0.0f, 0.0f, 0.0f, 0.0f};   // running max per owned column
    float* orow = map_h + (size_t)row * (NC * NC);
    const int h = row & (NH - 1);
    float* mrow = map_mask + (size_t)(row >> 9) * (NC * NC);

    for (int i = NC - 1; i >= 0; --i) {
        const float xi = xs[s * NC + i];
        // m_j(i) = x[i] if j==i ; max(m_j(i+1), x[i]) if j>i ; untouched if j<i
        m.x = (j0 + 0 >= i) ? ((j0 + 0 == i) ? xi : fmaxf(m.x, xi)) : m.x;
        m.y = (j0 + 1 >= i) ? ((j0 + 1 == i) ? xi : fmaxf(m.y, xi)) : m.y;
        m.z = (j0 + 2 >= i) ? ((j0 + 2 == i) ? xi : fmaxf(m.z, xi)) : m.z;
        m.w = (j0 + 3 >= i) ? ((j0 + 3 == i) ? xi : fmaxf(m.w, xi)) : m.w;

        f4 v;
        v.x = (j0 + 0 >= i) ? m.x : 0.0f;
        v.y = (j0 + 1 >= i) ? m.y : 0.0f;
        v.z = (j0 + 2 >= i) ? m.z : 0.0f;
        v.w = (j0 + 3 >= i) ? m.w : 0.0f;
        __builtin_nontemporal_store(v, (f4*)(orow + i * NC + j0));   // streaming: keep L2 clean

        if (h == 0) {
            f4 mk;
            mk.x = (j0 + 0 >= i) ? 1.0f : 0.0f;
            mk.y = (j0 + 1 >= i) ? 1.0f : 0.0f;
            mk.z = (j0 + 2 >= i) ? 1.0f : 0.0f;
            mk.w = (j0 + 3 >= i) ? 1.0f : 0.0f;
            __builtin_nontemporal_store(mk, (f4*)(mrow + i * NC + j0));
        }
    }
}

// ---------------------------------------------------------------------------
// Kernel 2: props_h[b,p,h] = max(x[b,h,s_p .. e_p-1]) via LDS sparse table.
// Block = (b, 32-wide h tile). st[k][c][hl]: lanes stride hl -> conflict-free.
// Props indices are wave-uniform -> scalar loads. Output NT, 128B per wave.
// ---------------------------------------------------------------------------
__global__ __launch_bounds__(256)
void dpm_props_kernel(const float* __restrict__ x,
                      const int* __restrict__ props,
                      float* __restrict__ out1) {
    __shared__ float st[7][NC][32];        // 56 KB
    const int t  = threadIdx.x;
    const int b  = blockIdx.x >> 4;
    const int h0 = (blockIdx.x & 15) * 32;
    const int hl = t & 31;
    const int cg = t >> 5;                 // 0..7 (wave id)
    const int c0 = cg * 8;

    // Load + transpose: x[b][h0+hl][c0..c0+7] -> st[0][c][hl]
    const f4* xp = (const f4*)(x + (size_t)(b * NH + h0 + hl) * NC + c0);
    f4 a = xp[0], c = xp[1];
    st[0][c0 + 0][hl] = a.x;  st[0][c0 + 1][hl] = a.y;
    st[0][c0 + 2][hl] = a.z;  st[0][c0 + 3][hl] = a.w;
    st[0][c0 + 4][hl] = c.x;  st[0][c0 + 5][hl] = c.y;
    st[0][c0 + 6][hl] = c.z;  st[0][c0 + 7][hl] = c.w;
    __syncthreads();

    // Sparse-table levels: st[k][c] = max over x[c .. c+2^k-1]
    for (int k = 1; k < 7; ++k) {
        const int half = 1 << (k - 1);
        for (int idx = t; idx < NC * 32; idx += 256) {
            const int cc = idx >> 5, hh = idx & 31;
            int c2 = cc + half; c2 = (c2 > NC - 1) ? (NC - 1) : c2;
            st[k][cc][hh] = fmaxf(st[k - 1][cc][hh], st[k - 1][c2][hh]);
        }
        __syncthreads();
    }

    // Queries: each wave owns one p-slot, lanes cover 32 consecutive h.
    float* obase = out1 + (size_t)b * (NP * NH) + h0 + hl;
    for (int p = cg; p < NP; p += 8) {
        const int sIdx = props[2 * p];          // wave-uniform -> s_load
        const int eIdx = props[2 * p + 1];
        const int len  = eIdx - sIdx;           // in [1, 64]
        const int k    = 31 - __builtin_clz(len);
        const float v  = fmaxf(st[k][sIdx][hl], st[k][eIdx - (1 << k)][hl]);
        __builtin_nontemporal_store(v, obase + (size_t)p * NH);
    }
}

extern "C" void kernel_launch(void* const* d_in, const int* in_sizes, int n_in,
                              void* d_out, int out_size, void* d_ws, size_t ws_size,
                              hipStream_t stream) {
    (void)in_sizes; (void)n_in; (void)out_size; (void)d_ws; (void)ws_size;
    const float* x     = (const float*)d_in[0];
    const int*   props = (const int*)d_in[1];
    float* out = (float*)d_out;

    dpm_map_kernel<<<(NB * NH) / 4, 64, 0, stream>>>(x, out + OUT2_OFF, out + OUT3_OFF);
    dpm_props_kernel<<<NB * (NH / 32), 256, 0, stream>>>(x, props, out);
}